// LSTMTagger_41145786696135
// MI455X (gfx1250) — compile-verified
//
#include <hip/hip_runtime.h>
#include <math.h>

#define SEQ      4096
#define WLEN     12
#define CHAR_DIM 64
#define CHAR_HID 128
#define NDIM     512
#define NHID     512
#define NTAG     64
#define FEAT     (NDIM + CHAR_HID)   // 640
#define CG       (4 * CHAR_HID)      // 512 char-lstm gates
#define GW       (4 * NHID)          // 2048 word-lstm gates
#define WLSTM_WGS 64

typedef __attribute__((ext_vector_type(2))) float v2f;
typedef __attribute__((ext_vector_type(8))) float v8f;

__device__ __forceinline__ v8f wmma_f32(v2f a, v2f b, v8f c) {
    // D = A(16x4,f32) * B(4x16,f32) + C(16x16,f32)
    return __builtin_amdgcn_wmma_f32_16x16x4_f32(
        /*neg_a=*/false, a, /*neg_b=*/false, b,
        /*c_mod=*/(short)0, c, /*reuse_a=*/false, /*reuse_b=*/false);
}

__device__ __forceinline__ float sigmoidf_(float x) {
    return 1.0f / (1.0f + __expf(-x));
}

// ---------------------------------------------------------------------------
// Async global->LDS copy (CDNA5 GLOBAL_LOAD_ASYNC_TO_LDS_B128, GVS mode).
// lds_byte_addr: LDS byte address (low 32 bits of flat shared pointer).
// gbl_byte_off : per-lane unsigned byte offset added to the SGPR base.
// Tracked with ASYNCcnt; drain with s_wait_asynccnt 0 then barrier.
// ---------------------------------------------------------------------------
__device__ __forceinline__ void async_copy_b128(unsigned lds_byte_addr,
                                                unsigned gbl_byte_off,
                                                const void* sbase) {
    asm volatile("global_load_async_to_lds_b128 %0, %1, %2"
                 :: "v"(lds_byte_addr), "v"(gbl_byte_off), "s"(sbase)
                 : "memory");
}
__device__ __forceinline__ void wait_async0() {
    asm volatile("s_wait_asynccnt 0x0" ::: "memory");
}
__device__ __forceinline__ unsigned lds_addr_of(const void* p) {
    return (unsigned)(size_t)p;   // aperture low 32 bits == LDS byte offset
}

// ---------------------------------------------------------------------------
// Kernel 0: deterministic workspace init (h ping-pong buffer + sync counter)
// ---------------------------------------------------------------------------
__global__ void init_kernel(float* hbuf, unsigned* syncc) {
    int i = threadIdx.x + blockIdx.x * blockDim.x;
    if (i < 2 * NHID) hbuf[i] = 0.0f;
    if (i == 0) *syncc = 0u;
}

// ---------------------------------------------------------------------------
// Kernel 1: char LSTM over 12 chars for a block of 16 words; emits feat rows
//           feat[word] = [ word_emb[word_idx[word]] (512) | c_final (128) ]
// Block: 256 threads (8 waves). Wave w computes gates [w*64, w*64+64).
// ---------------------------------------------------------------------------
__global__ __launch_bounds__(256)
void char_lstm_kernel(const int* __restrict__ char_idx,
                      const float* __restrict__ char_emb,
                      const float* __restrict__ cWih,   // [CG][CHAR_DIM]
                      const float* __restrict__ cWhh,   // [CG][CHAR_HID]
                      const float* __restrict__ cbih,
                      const float* __restrict__ cbhh,
                      const int* __restrict__ word_idx,
                      const float* __restrict__ word_emb,
                      float* __restrict__ feat) {
    const int word0 = blockIdx.x * 16;
    const int tid  = threadIdx.x;
    const int lane = tid & 31;
    const int wave = tid >> 5;

    __shared__ float s_x[16 * CHAR_DIM];   //  4 KB: x_t tile
    __shared__ float s_h[16 * CHAR_HID];   //  8 KB
    __shared__ float s_c[16 * CHAR_HID];   //  8 KB
    __shared__ float s_g[16 * CG];         // 32 KB: gates

    for (int i = tid; i < 16 * CHAR_HID; i += 256) { s_h[i] = 0.0f; s_c[i] = 0.0f; }
    __syncthreads();

    const int mrow = lane & 15;            // M (words) / N (gates) sub-index
    const int koff = (lane < 16) ? 0 : 2;  // K half split across lane halves

    for (int t = 0; t < WLEN; ++t) {
        // gather x_t = char_emb[char_idx[word, t]]  (16 x 64)
        for (int i = tid; i < 16 * CHAR_DIM; i += 256) {
            int m = i >> 6, d = i & 63;
            int ci = char_idx[(word0 + m) * WLEN + t];
            s_x[i] = char_emb[ci * CHAR_DIM + d];
        }
        __syncthreads();

        #pragma unroll
        for (int nt = 0; nt < 4; ++nt) {
            const int n0 = wave * 64 + nt * 16;
            const int n  = n0 + mrow;      // this lane's gate row (B column)
            v8f acc = {};
            // x_t @ cWih^T : K = 64
            #pragma unroll
            for (int k0 = 0; k0 < CHAR_DIM; k0 += 4) {
                v2f a, b;
                a.x = s_x[mrow * CHAR_DIM + k0 + koff];
                a.y = s_x[mrow * CHAR_DIM + k0 + koff + 1];
                b.x = cWih[n * CHAR_DIM + k0 + koff];
                b.y = cWih[n * CHAR_DIM + k0 + koff + 1];
                acc = wmma_f32(a, b, acc);
            }
            // h @ cWhh^T : K = 128
            #pragma unroll
            for (int k0 = 0; k0 < CHAR_HID; k0 += 4) {
                v2f a, b;
                a.x = s_h[mrow * CHAR_HID + k0 + koff];
                a.y = s_h[mrow * CHAR_HID + k0 + koff + 1];
                b.x = cWhh[n * CHAR_HID + k0 + koff];
                b.y = cWhh[n * CHAR_HID + k0 + koff + 1];
                acc = wmma_f32(a, b, acc);
            }
            const float bias = cbih[n] + cbhh[n];
            #pragma unroll
            for (int v = 0; v < 8; ++v) {
                int m = v + ((lane < 16) ? 0 : 8);
                s_g[m * CG + n] = acc[v] + bias;
            }
        }
        __syncthreads();

        // elementwise LSTM cell update (torch gate order i,f,g,o)
        for (int i = tid; i < 16 * CHAR_HID; i += 256) {
            int m = i >> 7, j = i & 127;
            float ig = s_g[m * CG + j];
            float fg = s_g[m * CG + CHAR_HID + j];
            float gg = s_g[m * CG + 2 * CHAR_HID + j];
            float og = s_g[m * CG + 3 * CHAR_HID + j];
            float c = sigmoidf_(fg) * s_c[i] + sigmoidf_(ig) * tanhf(gg);
            float h = sigmoidf_(og) * tanhf(c);
            s_c[i] = c;
            s_h[i] = h;
        }
        __syncthreads();
    }

    // feat = [word_emb | c_final]
    for (int i = tid; i < 16 * NDIM; i += 256) {
        int m = i >> 9, d = i & 511;
        int widx = word_idx[word0 + m];
        feat[(size_t)(word0 + m) * FEAT + d] = word_emb[(size_t)widx * NDIM + d];
    }
    for (int i = tid; i < 16 * CHAR_HID; i += 256) {
        int m = i >> 7, j = i & 127;
        feat[(size_t)(word0 + m) * FEAT + NDIM + j] = s_c[i];
    }
}

// ---------------------------------------------------------------------------
// Kernel 2: Gx = feat @ wWih^T + (wbih + wbhh)   [4096,640]x[640,2048]
// Block 256 (8 waves); wave -> one 16x16 tile; WG tile 16(M) x 128(N).
// A-tile (40 KB) staged via async global->LDS b128 copies (ASYNCcnt path).
// ---------------------------------------------------------------------------
__global__ __launch_bounds__(256)
void wgemm_gates_kernel(const float* __restrict__ feat,
                        const float* __restrict__ wWih,   // [GW][FEAT]
                        const float* __restrict__ wbih,
                        const float* __restrict__ wbhh,
                        float* __restrict__ Gx) {
    const int m0  = blockIdx.x * 16;
    const int tid = threadIdx.x, lane = tid & 31, wave = tid >> 5;

    __shared__ float s_a[16 * FEAT];   // 40 KB
    {
        const float* gsrc = feat + (size_t)m0 * FEAT;   // 16 contiguous rows
        const unsigned lbase = lds_addr_of(&s_a[0]);
        #pragma unroll
        for (int it = 0; it < (16 * FEAT * 4) / (256 * 16); ++it) {   // 10
            unsigned off = (unsigned)(it * 4096 + tid * 16);
            async_copy_b128(lbase + off, off, gsrc);
        }
        wait_async0();
    }
    __syncthreads();

    const int mrow = lane & 15;
    const int koff = (lane < 16) ? 0 : 2;
    const int n = blockIdx.y * 128 + wave * 16 + mrow;

    v8f acc = {};
    for (int k0 = 0; k0 < FEAT; k0 += 4) {
        v2f a, b;
        a.x = s_a[mrow * FEAT + k0 + koff];
        a.y = s_a[mrow * FEAT + k0 + koff + 1];
        b.x = wWih[(size_t)n * FEAT + k0 + koff];
        b.y = wWih[(size_t)n * FEAT + k0 + koff + 1];
        acc = wmma_f32(a, b, acc);
    }
    const float bias = wbih[n] + wbhh[n];
    #pragma unroll
    for (int v = 0; v < 8; ++v) {
        int m = v + ((lane < 16) ? 0 : 8);
        Gx[(size_t)(m0 + m) * GW + n] = acc[v] + bias;
    }
}

// ---------------------------------------------------------------------------
// Kernel 3: sequential word LSTM (batch=1, 4096 steps), persistent kernel.
// 64 WGs; WG w owns hidden slice j in [8w, 8w+8) -> gate rows {j,512+j,1024+j,1536+j}.
// Its 32 wWhh rows (64 KB) are cached in LDS once. Grid barrier = cumulative
// atomic counter; h ping-pongs between two global buffers.
// ---------------------------------------------------------------------------
__global__ __launch_bounds__(256)
void word_lstm_kernel(const float* __restrict__ Gx,
                      const float* __restrict__ wWhh,   // [GW][NHID]
                      float* __restrict__ hs,
                      float* __restrict__ hbuf,         // [2][NHID]
                      unsigned* __restrict__ syncc) {
    const int w   = blockIdx.x;   // 0..63
    const int tid = threadIdx.x;

    __shared__ float s_w[32 * NHID];   // 64 KB weight cache
    __shared__ float s_hv[NHID];
    __shared__ float s_part[256];
    __shared__ float s_gate[32];
    __shared__ float s_cj[8];

    for (int i = tid; i < 32 * NHID; i += 256) {
        int gi = i >> 9, k = i & 511;
        int row = (gi >> 3) * NHID + w * 8 + (gi & 7);
        s_w[i] = wWhh[(size_t)row * NHID + k];
    }
    if (tid < 8) s_cj[tid] = 0.0f;
    __syncthreads();

    const int gi = tid >> 3;                       // gate 0..31 within WG
    const int p  = tid & 7;                        // partial 0..7

    for (int t = 0; t < SEQ; ++t) {
        const float* hin  = hbuf + (t & 1) * NHID;
        float*       hout = hbuf + ((t + 1) & 1) * NHID;

        for (int i = tid; i < NHID; i += 256) s_hv[i] = hin[i];
        __syncthreads();

        float partial = 0.0f;
        const float* wr = &s_w[gi * NHID + p * 64];
        const float* hv = &s_hv[p * 64];
        #pragma unroll 8
        for (int k = 0; k < 64; ++k) partial = fmaf(wr[k], hv[k], partial);
        s_part[tid] = partial;
        __syncthreads();

        if (tid < 32) {
            float s = 0.0f;
            #pragma unroll
            for (int q = 0; q < 8; ++q) s += s_part[tid * 8 + q];
            int row = (tid >> 3) * NHID + w * 8 + (tid & 7);
            s_gate[tid] = s + Gx[(size_t)t * GW + row];
        }
        __syncthreads();

        if (tid < 8) {
            float ig = s_gate[tid],      fg = s_gate[8 + tid];
            float gg = s_gate[16 + tid], og = s_gate[24 + tid];
            float c = sigmoidf_(fg) * s_cj[tid] + sigmoidf_(ig) * tanhf(gg);
            float h = sigmoidf_(og) * tanhf(c);
            s_cj[tid] = c;
            hout[w * 8 + tid] = h;
            hs[(size_t)t * NHID + w * 8 + tid] = h;
        }
        __threadfence();
        __syncthreads();

        if (tid == 0) {
            __hip_atomic_fetch_add(syncc, 1u, __ATOMIC_RELEASE, __HIP_MEMORY_SCOPE_AGENT);
            const unsigned target = (unsigned)(t + 1) * WLSTM_WGS;
            while (__hip_atomic_load(syncc, __ATOMIC_ACQUIRE, __HIP_MEMORY_SCOPE_AGENT) < target)
                __builtin_amdgcn_s_sleep(1);
        }
        __syncthreads();
    }
}

// ---------------------------------------------------------------------------
// Kernel 4: logits = hs @ linW^T + linb, then per-row log_softmax over 64 tags.
// Block 128 (4 waves); wave -> one 16x16 N-tile (N=64 total).
// hs tile (32 KB) staged via async global->LDS b128 copies.
// ---------------------------------------------------------------------------
__global__ __launch_bounds__(128)
void proj_softmax_kernel(const float* __restrict__ hs,
                         const float* __restrict__ linW,   // [NTAG][NHID]
                         const float* __restrict__ linb,
                         float* __restrict__ out) {
    const int m0  = blockIdx.x * 16;
    const int tid = threadIdx.x, lane = tid & 31, wave = tid >> 5;

    __shared__ float s_a[16 * NHID];   // 32 KB
    __shared__ float s_l[16 * NTAG];   //  4 KB

    {
        const float* gsrc = hs + (size_t)m0 * NHID;     // 16 contiguous rows
        const unsigned lbase = lds_addr_of(&s_a[0]);
        #pragma unroll
        for (int it = 0; it < (16 * NHID * 4) / (128 * 16); ++it) {   // 16
            unsigned off = (unsigned)(it * 2048 + tid * 16);
            async_copy_b128(lbase + off, off, gsrc);
        }
        wait_async0();
    }
    __syncthreads();

    const int mrow = lane & 15;
    const int koff = (lane < 16) ? 0 : 2;
    const int n = wave * 16 + mrow;

    v8f acc = {};
    for (int k0 = 0; k0 < NHID; k0 += 4) {
        v2f a, b;
        a.x = s_a[mrow * NHID + k0 + koff];
        a.y = s_a[mrow * NHID + k0 + koff + 1];
        b.x = linW[(size_t)n * NHID + k0 + koff];
        b.y = linW[(size_t)n * NHID + k0 + koff + 1];
        acc = wmma_f32(a, b, acc);
    }
    const float bias = linb[n];
    #pragma unroll
    for (int v = 0; v < 8; ++v) {
        int m = v + ((lane < 16) ? 0 : 8);
        s_l[m * NTAG + n] = acc[v] + bias;
    }
    __syncthreads();

    if (tid < 16) {
        float mx = -1e30f;
        for (int j = 0; j < NTAG; ++j) mx = fmaxf(mx, s_l[tid * NTAG + j]);
        float se = 0.0f;
        for (int j = 0; j < NTAG; ++j) se += __expf(s_l[tid * NTAG + j] - mx);
        const float lse = mx + __logf(se);
        for (int j = 0; j < NTAG; ++j)
            out[(size_t)(m0 + tid) * NTAG + j] = s_l[tid * NTAG + j] - lse;
    }
}

// ---------------------------------------------------------------------------
extern "C" void kernel_launch(void* const* d_in, const int* in_sizes, int n_in,
                              void* d_out, int out_size, void* d_ws, size_t ws_size,
                              hipStream_t stream) {
    const int*   word_idx = (const int*)  d_in[0];
    const int*   char_idx = (const int*)  d_in[1];
    const float* word_emb = (const float*)d_in[2];
    const float* char_emb = (const float*)d_in[3];
    const float* cWih     = (const float*)d_in[4];
    const float* cWhh     = (const float*)d_in[5];
    const float* cbih     = (const float*)d_in[6];
    const float* cbhh     = (const float*)d_in[7];
    const float* wWih     = (const float*)d_in[8];
    const float* wWhh     = (const float*)d_in[9];
    const float* wbih     = (const float*)d_in[10];
    const float* wbhh     = (const float*)d_in[11];
    const float* linW     = (const float*)d_in[12];
    const float* linb     = (const float*)d_in[13];
    (void)in_sizes; (void)n_in; (void)out_size; (void)ws_size;

    float*    feat  = (float*)d_ws;                          // [SEQ][FEAT]
    float*    Gx    = feat + (size_t)SEQ * FEAT;             // [SEQ][GW]
    float*    hs    = Gx   + (size_t)SEQ * GW;               // [SEQ][NHID]
    float*    hbuf  = hs   + (size_t)SEQ * NHID;             // [2][NHID]
    unsigned* syncc = (unsigned*)(hbuf + 2 * NHID);

    init_kernel<<<4, 256, 0, stream>>>(hbuf, syncc);

    char_lstm_kernel<<<SEQ / 16, 256, 0, stream>>>(
        char_idx, char_emb, cWih, cWhh, cbih, cbhh, word_idx, word_emb, feat);

    dim3 gB(SEQ / 16, GW / 128);
    wgemm_gates_kernel<<<gB, 256, 0, stream>>>(feat, wWih, wbih, wbhh, Gx);

    word_lstm_kernel<<<WLSTM_WGS, 256, 0, stream>>>(Gx, wWhh, hs, hbuf, syncc);

    proj_softmax_kernel<<<SEQ / 16, 128, 0, stream>>>(hs, linW, linb, (float*)d_out);
}